// SSD_R34_41283225649625
// MI455X (gfx1250) — compile-verified
//
#include <hip/hip_runtime.h>
#include <hip/hip_bf16.h>

// ---------------- problem constants (match reference) ----------------
#define Bsz      32
#define Ccls     81
#define Nbx      15130
#define Ksel     200
#define MAXOUT   200
#define NN2      ((Ccls - 1) * Ksel)   // 16000 candidates per image
#define SCORE_THR 0.05f
#define CRITERIA  0.45f
#define NEG_INF  (-1e30f)
#define THREADS  256
#define BN       (Bsz * Nbx)
#define NGROUPS  (BN / 16)             // 30260 row-groups of 16 for WMMA softmax
#define CHUNK3   6144                  // floats staged per TDM chunk (24 KB), kernel 3
#define CHUNK4   6144                  // floats staged per async chunk (24 KB), kernel 4

typedef float v2f __attribute__((ext_vector_type(2)));
typedef float v8f __attribute__((ext_vector_type(8)));
typedef unsigned int v4u __attribute__((ext_vector_type(4)));
typedef int v4i __attribute__((ext_vector_type(4)));
typedef int v8i __attribute__((ext_vector_type(8)));

// ---------------- helpers: order-preserving float<->uint ----------------
__device__ __forceinline__ unsigned int f32_ord(float f) {
  unsigned int u = __float_as_uint(f);
  return (u & 0x80000000u) ? ~u : (u | 0x80000000u);
}
__device__ __forceinline__ float ord_f32(unsigned int u) {
  unsigned int b = (u & 0x80000000u) ? (u & 0x7FFFFFFFu) : ~u;
  return __uint_as_float(b);
}
__device__ __forceinline__ unsigned long long make_key(float v, unsigned int idx) {
  // descending by value; ties broken by LOWER index first (stable top_k)
  return ((unsigned long long)f32_ord(v) << 32) |
         (unsigned long long)(0xFFFFFFFFu - idx);
}

// ---------------- CDNA5 async global->LDS (ASYNCcnt path) ----------------
__device__ __forceinline__ unsigned lds_byte_off(const void* p) {
  return (unsigned)(unsigned long long)(uintptr_t)p;   // AS3 offset = low 32 bits
}
__device__ __forceinline__ void async_ld_b128(unsigned lds_off, const void* gptr) {
  asm volatile("global_load_async_to_lds_b128 %0, %1, off"
               :: "v"(lds_off), "v"((unsigned long long)(uintptr_t)gptr)
               : "memory");
}
__device__ __forceinline__ void wait_async0() {
  asm volatile("s_wait_asynccnt 0x0" ::: "memory");
}

// ---------------- CDNA5 Tensor Data Mover: 1-D row DMA global->LDS ----------------
// Builds a D# for a contiguous nelem*4-byte copy: data_size=4B, tensor_dim0=tile_dim0=nelem,
// all higher dims/tiles = 1, strides benign. Tracked by TENSORcnt.
__device__ __forceinline__ void tdm_load_row(unsigned lds_off, const void* gsrc, int nelem) {
  unsigned long long ga = (unsigned long long)(uintptr_t)gsrc;
  unsigned un = (unsigned)nelem;
  v4u g0;
  g0[0] = 1u;                                            // count=1, user descriptor
  g0[1] = lds_off;                                       // lds_addr (bytes)
  g0[2] = (unsigned)(ga & 0xFFFFFFFFull);                // global_addr[31:0]
  g0[3] = (unsigned)((ga >> 32) & 0x01FFFFFFull)         // global_addr[56:32]
          | 0x80000000u;                                 // type=2 ("image")
  v8i g1;
  g1[0] = 0x00020000;                                    // data_size=4B, mask=0, flags=0
  g1[1] = (int)((un & 0xFFFFu) << 16);                   // tensor_dim0[15:0]
  g1[2] = (int)(((un >> 16) & 0xFFFFu) | (1u << 16));    // tensor_dim0[31:16] | tensor_dim1=1
  g1[3] = (int)((un & 0xFFFFu) << 16);                   // tensor_dim1 hi=0 | tile_dim0=nelem
  g1[4] = 0x00010001;                                    // tile_dim1=1 | tile_dim2=1
  g1[5] = (int)un;                                       // tensor_dim0_stride[31:0]
  g1[6] = 0;                                             // stride0 hi | stride1 lo
  g1[7] = 0;                                             // tensor_dim1_stride hi
  v4i g2; g2[0] = 1; g2[1] = 1; g2[2] = 0; g2[3] = 0x00010000; // dim2=1,dim3=1,stride2=0,tile_dim3=1
  v4i g3; g3[0] = 0; g3[1] = 0x00010000; g3[2] = 0x00010000; g3[3] = 0; // stride3=0,dim4=1,tile_dim4=1
#if __clang_major__ >= 23
  v8i gpad = {0, 0, 0, 0, 0, 0, 0, 0};
  __builtin_amdgcn_tensor_load_to_lds(g0, g1, g2, g3, gpad, 0);
#else
  __builtin_amdgcn_tensor_load_to_lds(g0, g1, g2, g3, 0);
#endif
}

// ---------------- kernel 1: softmax max & denom (WMMA f32 row-sums) ----------------
// Each wave handles 16 rows (one (b,n) per lane pair). Row max via lane-split +
// shfl_xor(16); exp-sum via V_WMMA_F32_16X16X4_F32 with B=ones (full f32, RNE):
// A[16x4] layout = lanes 0-15 hold K=0,1 / lanes 16-31 hold K=2,3 per ISA 7.12.2.
__global__ void __launch_bounds__(THREADS)
k_softmax_stats(const float* __restrict__ scores,
                float* __restrict__ rowMax, float* __restrict__ rowDen) {
  const int lane = threadIdx.x & 31;
  const int gw = blockIdx.x * (THREADS / 32) + (threadIdx.x >> 5);
  if (gw >= NGROUPS) return;                 // wave-uniform exit: EXEC stays all-1s
  const int r = gw * 16 + (lane & 15);       // row for this lane (pair with lane^16)
  const int b = r / Nbx, n = r % Nbx;
  const float* prow = scores + (size_t)b * Ccls * Nbx + n;

  // row max: lanes <16 scan classes [0,41), lanes >=16 scan [41,81), combine
  float m = -__builtin_inff();
  const int c0 = (lane < 16) ? 0 : 41;
  const int c1 = (lane < 16) ? 41 : Ccls;
  for (int c = c0; c < c1; ++c) {
    __builtin_prefetch(prow + (size_t)(c + 8) * Nbx, 0, 1);
    m = fmaxf(m, prow[(size_t)c * Nbx]);
  }
  m = fmaxf(m, __shfl_xor(m, 16, 32));

  // exp-sum: 20 WMMA K-chunks of 4 classes; class 80 handled scalar
  v8f acc = {0.f, 0.f, 0.f, 0.f, 0.f, 0.f, 0.f, 0.f};
  v2f ones; ones[0] = 1.0f; ones[1] = 1.0f;
  const int koff = (lane < 16) ? 0 : 2;
  for (int cb = 0; cb + 4 <= Ccls; cb += 4) {
    v2f a;
    a[0] = expf(prow[(size_t)(cb + koff + 0) * Nbx] - m);
    a[1] = expf(prow[(size_t)(cb + koff + 1) * Nbx] - m);
    acc = __builtin_amdgcn_wmma_f32_16x16x4_f32(false, a, false, ones,
                                                (short)0, acc, false, false);
  }
  float e80 = expf(prow[(size_t)(Ccls - 1) * Nbx] - m);

  if (lane < 16) rowMax[r] = m;
  // D layout: lanes 0-15 hold rows gw*16+0..7 in acc[0..7]; lanes 16-31 rows 8..15.
  float e80w = __shfl(e80, (lane < 16) ? lane : (lane - 8), 32);
  if ((lane & 15) < 8) {
    const int j = lane & 7;
    float sv = 0.0f;
#pragma unroll
    for (int t = 0; t < 8; ++t) if (j == t) sv = acc[t];
    const int row = gw * 16 + ((lane < 16) ? j : (8 + j));
    rowDen[row] = sv + e80w;
  }
}

// ---------------- kernel 2: box decode to ltrb ----------------
__global__ void __launch_bounds__(THREADS)
k_decode(const float* __restrict__ bboxes, const float* __restrict__ dboxes,
         float* __restrict__ ltrb) {
  int g = blockIdx.x * blockDim.x + threadIdx.x;
  if (g >= BN) return;
  int b = g / Nbx, n = g % Nbx;
  const float* base = bboxes + (size_t)b * 4 * Nbx + n;
  float bx = base[0];
  float by = base[(size_t)Nbx];
  float bw = base[2 * (size_t)Nbx];
  float bh = base[3 * (size_t)Nbx];
  float dcx = dboxes[n * 4 + 0], dcy = dboxes[n * 4 + 1];
  float dw  = dboxes[n * 4 + 2], dh  = dboxes[n * 4 + 3];
  float cx = 0.1f * bx * dw + dcx;
  float cy = 0.1f * by * dh + dcy;
  float ww = expf(0.2f * bw) * dw;
  float hh = expf(0.2f * bh) * dh;
  float4 o;
  o.x = cx - 0.5f * ww;
  o.y = cy - 0.5f * hh;
  o.z = cx + 0.5f * ww;
  o.w = cy + 0.5f * hh;
  ((float4*)ltrb)[g] = o;
}

// -------- exact top-K via LDS radix-256 select + bitonic sort --------
__device__ __forceinline__ void topk_select_sort(
    unsigned long long* kv, int nn, int kwant,
    unsigned long long* topbuf,      // [THREADS]
    unsigned int* hist,              // [THREADS]
    unsigned long long* sPref,       // [2]: bits, mask
    int* sKneed, unsigned int* sCnt, int tid) {
  if (tid == 0) { sPref[0] = 0ull; sPref[1] = 0ull; *sKneed = kwant; }
  __syncthreads();
  for (int shift = 56; shift >= 0; shift -= 8) {
    hist[tid] = 0u;
    __syncthreads();
    unsigned long long pb = sPref[0], pm = sPref[1];
    for (int i = tid; i < nn; i += THREADS) {
      unsigned long long key = kv[i];
      if ((key & pm) == pb)
        atomicAdd(&hist[(unsigned int)(key >> shift) & 0xFFu], 1u);
    }
    __syncthreads();
    if (tid == 0) {
      int need = *sKneed;
      unsigned int cum = 0;
      int bsel = 0;
      for (int bb = 255; bb >= 0; --bb) {
        unsigned int h = hist[bb];
        if (cum + h >= (unsigned int)need) { bsel = bb; break; }
        cum += h;
      }
      *sKneed = need - (int)cum;
      sPref[0] |= ((unsigned long long)bsel) << shift;
      sPref[1] |= (0xFFull << shift);
    }
    __syncthreads();
  }
  unsigned long long T = sPref[0];  // kwant-th largest key (keys are unique)
  topbuf[tid] = 0ull;
  if (tid == 0) *sCnt = 0u;
  __syncthreads();
  for (int i = tid; i < nn; i += THREADS) {
    unsigned long long key = kv[i];
    if (key >= T) {
      unsigned int p = atomicAdd(sCnt, 1u);
      if (p < (unsigned int)THREADS) topbuf[p] = key;
    }
  }
  __syncthreads();
  // bitonic sort of 256 keys, descending
  for (unsigned int kk = 2; kk <= THREADS; kk <<= 1) {
    for (unsigned int j = kk >> 1; j > 0; j >>= 1) {
      unsigned int i = (unsigned int)tid;
      unsigned int ixj = i ^ j;
      if (ixj > i) {
        unsigned long long a = topbuf[i], c = topbuf[ixj];
        bool desc = ((i & kk) == 0u);
        if (desc ? (a < c) : (a > c)) { topbuf[i] = c; topbuf[ixj] = a; }
      }
      __syncthreads();
    }
  }
}

// ---------------- kernel 3: per-(image,class) top-K + greedy NMS ----------------
__global__ void __launch_bounds__(THREADS)
k_class_nms(const float* __restrict__ scores, const float* __restrict__ rowMax,
            const float* __restrict__ rowDen, const float* __restrict__ ltrb,
            float* __restrict__ clsScores, float* __restrict__ clsBoxes) {
  __shared__ unsigned long long kv[Nbx];                        // ~121 KB
  __shared__ __attribute__((aligned(16))) float stage[CHUNK3];  // 24 KB TDM staging
  __shared__ unsigned long long topbuf[THREADS];
  __shared__ unsigned int hist[THREADS];
  __shared__ unsigned long long sPref[2];
  __shared__ int sKneed;
  __shared__ unsigned int sCnt;
  __shared__ float bk[Ksel * 4];
  __shared__ float areaL[Ksel];
  __shared__ unsigned int keepL[Ksel];

  const int tid = threadIdx.x;
  const int blk = blockIdx.x;              // b * 80 + (c-1)
  const int b = blk / (Ccls - 1);
  const int c = blk % (Ccls - 1) + 1;      // skip background class 0

  const float* srow = scores + ((size_t)b * Ccls + c) * Nbx;
  const float* mrow = rowMax + (size_t)b * Nbx;
  const float* drow = rowDen + (size_t)b * Nbx;
  const unsigned stage_base = lds_byte_off(stage);

  // TDM-stage the (L2-resident) score row chunk-by-chunk, then build sort keys.
  for (int base = 0; base < Nbx; base += CHUNK3) {
    const int cnt = (Nbx - base < CHUNK3) ? (Nbx - base) : CHUNK3;
    if (tid < 32) {                        // wave 0 issues the tensor DMA
      tdm_load_row(stage_base, srow + base, cnt);
      __builtin_amdgcn_s_wait_tensorcnt((short)0);
    }
    __syncthreads();
    for (int i = tid; i < cnt; i += THREADS) {
      int n = base + i;
      float pr = expf(stage[i] - mrow[n]) / drow[n];
      float v = (pr > SCORE_THR) ? pr : NEG_INF;
      kv[n] = make_key(v, (unsigned int)n);
    }
    __syncthreads();   // stage reused next chunk
  }

  topk_select_sort(kv, Nbx, Ksel, topbuf, hist, sPref, &sKneed, &sCnt, tid);

  // gather candidate boxes + areas + initial keep flags
  if (tid < Ksel) {
    unsigned long long key = topbuf[tid];
    unsigned int n = 0xFFFFFFFFu - (unsigned int)(key & 0xFFFFFFFFu);
    float s = ord_f32((unsigned int)(key >> 32));
    const float* bp = ltrb + ((size_t)b * Nbx + n) * 4;
    float l = bp[0], t = bp[1], r = bp[2], bo = bp[3];
    bk[tid * 4 + 0] = l; bk[tid * 4 + 1] = t;
    bk[tid * 4 + 2] = r; bk[tid * 4 + 3] = bo;
    areaL[tid] = (r - l) * (bo - t);
    keepL[tid] = (s > NEG_INF) ? 1u : 0u;
  }
  __syncthreads();

  // greedy sequential NMS (faithful: intersection deltas NOT clamped at 0)
  for (int i = 0; i < Ksel; ++i) {
    if (tid > i && tid < Ksel) {
      if (keepL[i] && keepL[tid]) {
        float lt0 = fmaxf(bk[i * 4 + 0], bk[tid * 4 + 0]);
        float lt1 = fmaxf(bk[i * 4 + 1], bk[tid * 4 + 1]);
        float rb0 = fminf(bk[i * 4 + 2], bk[tid * 4 + 2]);
        float rb1 = fminf(bk[i * 4 + 3], bk[tid * 4 + 3]);
        float inter = (rb0 - lt0) * (rb1 - lt1);
        float iou = inter / (areaL[i] + areaL[tid] - inter);
        if (iou >= CRITERIA) keepL[tid] = 0u;
      }
    }
    __syncthreads();
  }

  if (tid < Ksel) {
    unsigned long long key = topbuf[tid];
    float s = ord_f32((unsigned int)(key >> 32));
    clsScores[(size_t)blk * Ksel + tid] = keepL[tid] ? s : NEG_INF;
    float4 o;
    o.x = bk[tid * 4 + 0]; o.y = bk[tid * 4 + 1];
    o.z = bk[tid * 4 + 2]; o.w = bk[tid * 4 + 3];
    ((float4*)clsBoxes)[(size_t)blk * Ksel + tid] = o;
  }
}

// ---------------- kernel 4: global top-200 per image, write outputs ----------------
__global__ void __launch_bounds__(THREADS)
k_final(const float* __restrict__ clsScores, const float* __restrict__ clsBoxes,
        float* __restrict__ outBoxes, int* __restrict__ outLabels,
        float* __restrict__ outScores) {
  __shared__ unsigned long long kv[NN2];                        // 128 KB
  __shared__ __attribute__((aligned(16))) float stage[CHUNK4];  // 24 KB async staging
  __shared__ unsigned long long topbuf[THREADS];
  __shared__ unsigned int hist[THREADS];
  __shared__ unsigned long long sPref[2];
  __shared__ int sKneed;
  __shared__ unsigned int sCnt;

  const int tid = threadIdx.x;
  const int b = blockIdx.x;

  const float* srow = clsScores + (size_t)b * NN2;          // 16B-aligned row
  const unsigned stage_base = lds_byte_off(stage);

  for (int base = 0; base < NN2; base += CHUNK4) {
    const int cnt = (NN2 - base < CHUNK4) ? (NN2 - base) : CHUNK4;
    const int ngrp = cnt >> 2;                              // exact: NN2 % 4 == 0
    for (int g = tid; g < ngrp; g += THREADS)
      async_ld_b128(stage_base + (unsigned)g * 16u, srow + base + g * 4);
    wait_async0();
    __syncthreads();
    for (int i = tid; i < cnt; i += THREADS)
      kv[base + i] = make_key(stage[i], (unsigned int)(base + i));
    __syncthreads();
  }

  topk_select_sort(kv, NN2, MAXOUT, topbuf, hist, sPref, &sKneed, &sCnt, tid);

  if (tid < MAXOUT) {
    unsigned long long key = topbuf[tid];
    float s = ord_f32((unsigned int)(key >> 32));
    unsigned int f = 0xFFFFFFFFu - (unsigned int)(key & 0xFFFFFFFFu);
    bool valid = s > NEG_INF;
    float4 box; box.x = box.y = box.z = box.w = 0.0f;
    int label = 0;
    if (valid) {
      box = ((const float4*)clsBoxes)[(size_t)b * NN2 + f];
      label = (int)(f / Ksel) + 1;
    }
    ((float4*)outBoxes)[(size_t)b * MAXOUT + tid] = box;
    outLabels[(size_t)b * MAXOUT + tid] = label;
    outScores[(size_t)b * MAXOUT + tid] = valid ? s : 0.0f;
  }
}

// ---------------- host launch ----------------
extern "C" void kernel_launch(void* const* d_in, const int* in_sizes, int n_in,
                              void* d_out, int out_size, void* d_ws, size_t ws_size,
                              hipStream_t stream) {
  (void)in_sizes; (void)n_in; (void)out_size; (void)ws_size;
  const float* bboxes = (const float*)d_in[0];   // [B,4,N]
  const float* scores = (const float*)d_in[1];   // [B,C,N]
  const float* dboxes = (const float*)d_in[2];   // [1,N,4]

  // workspace layout (floats); total ~21.9 MB
  float* ws        = (float*)d_ws;
  float* rowMax    = ws;                                   // BN
  float* rowDen    = ws + (size_t)BN;                      // BN
  float* ltrb      = ws + 2 * (size_t)BN;                  // BN*4
  float* clsScores = ws + 6 * (size_t)BN;                  // B*80*K
  float* clsBoxes  = clsScores + (size_t)Bsz * (Ccls - 1) * Ksel;  // B*80*K*4

  // output layout: boxes [B,200,4] f32 | labels [B,200] i32 | scores [B,200] f32
  float* outBoxes  = (float*)d_out;
  int*   outLabels = (int*)((float*)d_out + (size_t)Bsz * MAXOUT * 4);
  float* outScores = (float*)d_out + (size_t)Bsz * MAXOUT * 5;

  const int gridW = (NGROUPS + (THREADS / 32) - 1) / (THREADS / 32);
  const int grid1 = (BN + THREADS - 1) / THREADS;
  k_softmax_stats<<<gridW, THREADS, 0, stream>>>(scores, rowMax, rowDen);
  k_decode<<<grid1, THREADS, 0, stream>>>(bboxes, dboxes, ltrb);
  k_class_nms<<<Bsz * (Ccls - 1), THREADS, 0, stream>>>(scores, rowMax, rowDen,
                                                        ltrb, clsScores, clsBoxes);
  k_final<<<Bsz, THREADS, 0, stream>>>(clsScores, clsBoxes,
                                       outBoxes, outLabels, outScores);
}